// HyperPointNet_33603824123948
// MI455X (gfx1250) — compile-verified
//
#include <hip/hip_runtime.h>
#include <hip/hip_bf16.h>
#include <math.h>

// ---------------- constants ----------------
#define BATCH   16
#define NPTS    1024
#define KNN     20
#define CCURV   0.01f
#define SQRTC   0.1f
#define BALLEPS 4e-3f
#define NEDGE   (BATCH*NPTS*KNN)     // 327680
#define NPTS_T  (BATCH*NPTS)         // 16384

typedef __attribute__((ext_vector_type(2))) float v2f;
typedef __attribute__((ext_vector_type(8))) float v8f;

__device__ __forceinline__ float wred32(float v) {
  #pragma unroll
  for (int m = 1; m < 32; m <<= 1) v += __shfl_xor(v, m, 32);
  return v;
}

// ---------------- zero init (stats + pooled max buffers) ----------------
__global__ void zero2_kernel(float* __restrict__ a, int na, float* __restrict__ b, int nb) {
  int i = blockIdx.x * blockDim.x + threadIdx.x;
  if (i < na) a[i] = 0.f;
  if (i < nb) b[i] = 0.f;
}

// ---------------- e2p (expmap0 + ball projection), C=3 ----------------
__global__ void e2p3_kernel(const float* __restrict__ x, float* __restrict__ xp, int rows) {
  int r = blockIdx.x * blockDim.x + threadIdx.x;
  if (r >= rows) return;
  float a = x[r*3+0], b = x[r*3+1], c = x[r*3+2];
  float n = fmaxf(sqrtf(a*a + b*b + c*c), 1e-15f);
  float t = tanhf(SQRTC * n) / (SQRTC * n);
  float ya = a*t, yb = b*t, yc = c*t;
  float ny = fmaxf(n * t, 1e-15f);
  float maxn = (1.0f - BALLEPS) / SQRTC;
  float s = (ny > maxn) ? (maxn / ny) : 1.0f;
  xp[r*3+0] = ya*s; xp[r*3+1] = yb*s; xp[r*3+2] = yc*s;
}

// ---------------- e2p, C=64 (one wave per row) ----------------
__global__ void e2p64_kernel(const float* __restrict__ xin, int instride,
                             float* __restrict__ xp, int rows) {
  int wid  = (blockIdx.x * blockDim.x + threadIdx.x) >> 5;
  int lane = threadIdx.x & 31;
  if (wid >= rows) return;
  const float* src = xin + (size_t)wid * instride;
  float v0 = src[lane], v1 = src[lane + 32];
  float ss = wred32(v0*v0 + v1*v1);
  float n = fmaxf(sqrtf(ss), 1e-15f);
  float t = tanhf(SQRTC * n) / (SQRTC * n);
  float ny = fmaxf(n * t, 1e-15f);
  float maxn = (1.0f - BALLEPS) / SQRTC;
  float s = (ny > maxn) ? (maxn / ny) : 1.0f;
  xp[(size_t)wid*64 + lane]      = v0 * t * s;
  xp[(size_t)wid*64 + lane + 32] = v1 * t * s;
}

// ---------------- kNN (C=3, raw coords), exact top-20 ----------------
__global__ void knn1_kernel(const float* __restrict__ x, int* __restrict__ idxout) {
  __shared__ float vals[NPTS];
  __shared__ float rv[256];
  __shared__ int   ri[256];
  int bn = blockIdx.x;               // 0..16383
  int b = bn >> 10, n = bn & 1023;
  int t = threadIdx.x;
  const float* xb = x + (size_t)b * NPTS * 3;
  float qx = xb[n*3], qy = xb[n*3+1], qz = xb[n*3+2];
  float qq = qx*qx + qy*qy + qz*qz;
  for (int m = t; m < NPTS; m += 256) {
    float px = xb[m*3], py = xb[m*3+1], pz = xb[m*3+2];
    vals[m] = 2.f*(qx*px + qy*py + qz*pz) - qq - (px*px + py*py + pz*pz);
  }
  __syncthreads();
  for (int kk = 0; kk < KNN; kk++) {
    float bv = -3.4e38f; int bi = 1 << 30;
    for (int m = t; m < NPTS; m += 256) {
      float v = vals[m];
      if (v > bv || (v == bv && m < bi)) { bv = v; bi = m; }
    }
    rv[t] = bv; ri[t] = bi; __syncthreads();
    for (int s = 128; s > 0; s >>= 1) {
      if (t < s) {
        float ov = rv[t+s]; int oi = ri[t+s];
        if (ov > rv[t] || (ov == rv[t] && oi < ri[t])) { rv[t] = ov; ri[t] = oi; }
      }
      __syncthreads();
    }
    if (t == 0) { idxout[bn*KNN + kk] = ri[0]; vals[ri[0]] = -3.4e38f; }
    __syncthreads();
  }
}

// ---------------- kNN (C=64, x1 lives in xcat cols 0..63) ----------------
__global__ void knn2_kernel(const float* __restrict__ xcat, int* __restrict__ idxout) {
  __shared__ float vals[NPTS];
  __shared__ float qs[64];
  __shared__ float rv[256];
  __shared__ int   ri[256];
  int bn = blockIdx.x;
  int b = bn >> 10, n = bn & 1023;
  int t = threadIdx.x;
  const float* xb = xcat + (size_t)b * NPTS * 256;
  if (t < 64) qs[t] = xb[(size_t)n*256 + t];
  __syncthreads();
  float qq = 0.f;
  #pragma unroll 8
  for (int c = 0; c < 64; c++) qq += qs[c]*qs[c];
  for (int m = t; m < NPTS; m += 256) {
    const float* p = xb + (size_t)m*256;
    float dot = 0.f, pp = 0.f;
    #pragma unroll 8
    for (int c = 0; c < 64; c++) { float v = p[c]; dot += qs[c]*v; pp += v*v; }
    vals[m] = 2.f*dot - qq - pp;
  }
  __syncthreads();
  for (int kk = 0; kk < KNN; kk++) {
    float bv = -3.4e38f; int bi = 1 << 30;
    for (int m = t; m < NPTS; m += 256) {
      float v = vals[m];
      if (v > bv || (v == bv && m < bi)) { bv = v; bi = m; }
    }
    rv[t] = bv; ri[t] = bi; __syncthreads();
    for (int s = 128; s > 0; s >>= 1) {
      if (t < s) {
        float ov = rv[t+s]; int oi = ri[t+s];
        if (ov > rv[t] || (ov == rv[t] && oi < ri[t])) { rv[t] = ov; ri[t] = oi; }
      }
      __syncthreads();
    }
    if (t == 0) { idxout[bn*KNN + kk] = ri[0]; vals[ri[0]] = -3.4e38f; }
    __syncthreads();
  }
}

// ------------ mobius coefficients for layer2: u = pA*feat - pB*xc ------------
__global__ void mcoef2_kernel(const float* __restrict__ xp2, const int* __restrict__ idx2,
                              float2* __restrict__ coef) {
  int s = (blockIdx.x * blockDim.x + threadIdx.x) >> 5;   // one wave per edge
  int lane = threadIdx.x & 31;
  if (s >= NEDGE) return;
  int bn = s / KNN; int b = bn >> 10;
  int m = idx2[s];
  const float* feat = xp2 + ((size_t)(b*NPTS + m)) * 64;
  const float* xc   = xp2 + (size_t)bn * 64;
  float f0 = feat[lane], f1 = feat[lane+32];
  float c0 = xc[lane],   c1 = xc[lane+32];
  float x2  = wred32(f0*f0 + f1*f1);
  float y2  = wred32(c0*c0 + c1*c1);
  float dfx = wred32(f0*c0 + f1*c1);
  if (lane == 0) {
    float den = fmaxf(1.f - 2.f*CCURV*dfx + CCURV*CCURV*x2*y2, 1e-15f);
    float pA = (1.f - 2.f*CCURV*dfx + CCURV*y2) / den;
    float pB = (1.f - CCURV*x2) / den;
    coef[s] = make_float2(pA, pB);
  }
}

// ---------------- conv1 (64x6, tiny -> VALU), MODE 0=stats 1=apply ----------------
template<int MODE>
__global__ void conv1_kernel(const float* __restrict__ xp1, const int* __restrict__ idx1,
                             const float* __restrict__ w1,
                             const float* __restrict__ scale, const float* __restrict__ shift,
                             float* __restrict__ gsum, float* __restrict__ gsq,
                             float* __restrict__ xcat) {
  __shared__ float f6[KNN][6];
  int bn = blockIdx.x; int b = bn >> 10;
  int t = threadIdx.x;                    // 64 threads = 64 output channels
  if (t < KNN) {
    int m = idx1[bn*KNN + t];
    const float* fp = xp1 + ((size_t)(b*NPTS + m)) * 3;
    const float* xc = xp1 + (size_t)bn * 3;
    float fx = fp[0], fy = fp[1], fz = fp[2];
    float cx = xc[0], cy = xc[1], cz = xc[2];
    float x2  = fx*fx + fy*fy + fz*fz;
    float y2  = cx*cx + cy*cy + cz*cz;
    float dfx = fx*cx + fy*cy + fz*cz;
    float den = fmaxf(1.f - 2.f*CCURV*dfx + CCURV*CCURV*x2*y2, 1e-15f);
    float pA = (1.f - 2.f*CCURV*dfx + CCURV*y2) / den;
    float pB = (1.f - CCURV*x2) / den;
    f6[t][0] = pA*fx - pB*cx; f6[t][1] = pA*fy - pB*cy; f6[t][2] = pA*fz - pB*cz;
    f6[t][3] = cx; f6[t][4] = cy; f6[t][5] = cz;
  }
  __syncthreads();
  float w[6];
  #pragma unroll
  for (int j = 0; j < 6; j++) w[j] = w1[t*6 + j];
  float sc = 0.f, sh = 0.f;
  if (MODE == 1) { sc = scale[t]; sh = shift[t]; }
  float s = 0.f, q = 0.f, mx = 0.f;
  for (int kk = 0; kk < KNN; kk++) {
    float h = 0.f;
    #pragma unroll
    for (int j = 0; j < 6; j++) h += w[j] * f6[kk][j];
    if (MODE == 0) { s += h; q += h*h; }
    else           { mx = fmaxf(mx, fmaxf(0.f, h*sc + sh)); }
  }
  if (MODE == 0) { atomicAdd(&gsum[t], s); atomicAdd(&gsq[t], q); }
  else           { xcat[(size_t)bn*256 + t] = mx; }
}

// ---------------- BN finalize: scale/shift per channel ----------------
__global__ void bnfin_kernel(const float* __restrict__ gsum, const float* __restrict__ gsq,
                             const float* __restrict__ g, const float* __restrict__ bb,
                             float* __restrict__ scale, float* __restrict__ shift,
                             int C, float invcount) {
  int c = blockIdx.x * blockDim.x + threadIdx.x;
  if (c >= C) return;
  float mean = gsum[c] * invcount;
  float var  = gsq[c] * invcount - mean*mean;
  float sc = g[c] * rsqrtf(var + 1e-5f);
  scale[c] = sc; shift[c] = bb[c] - mean * sc;
}

// ---------------- conv2: WMMA f32, D(192 x 80cols) per WG (4 pts x 20 nbrs) ----------------
template<int MODE>
__global__ __launch_bounds__(256)
void conv2_kernel(const float* __restrict__ xp2, const int* __restrict__ idx2,
                  const float2* __restrict__ coef, const float* __restrict__ w2,
                  const float* __restrict__ scale, const float* __restrict__ shift,
                  float* __restrict__ gsum, float* __restrict__ gsq,
                  float* __restrict__ xcat) {
  __shared__ float  F[80][130];          // [col][k], padded to dodge bank conflicts
  __shared__ float  xcs[4][64];
  __shared__ int    colm[80];
  __shared__ float2 colab[80];
  __shared__ float  sumL[192], sqL[192];
  __shared__ int    xmaxI[4*192];
  int t = threadIdx.x;
  int p0 = blockIdx.x * 4;               // global point index
  int b = p0 >> 10;
  if (MODE == 0) { if (t < 192) { sumL[t] = 0.f; sqL[t] = 0.f; } }
  else           { for (int i = t; i < 4*192; i += 256) xmaxI[i] = 0; }
  { int pt = t >> 6, c = t & 63;
    xcs[pt][c] = xp2[((size_t)(p0 + pt))*64 + c]; }
  if (t < 80) {
    int pt = t / KNN, kk = t % KNN;
    int s = (p0 + pt)*KNN + kk;
    colm[t]  = idx2[s];
    colab[t] = coef[s];
  }
  __syncthreads();
  for (int i = t; i < 80*128; i += 256) {
    int col = i >> 7, c = i & 127;
    float v;
    if (c < 64) {
      float fv = xp2[((size_t)(b*NPTS + colm[col]))*64 + c];
      float2 ab = colab[col];
      v = ab.x * fv - ab.y * xcs[col/KNN][c];
    } else {
      v = xcs[col/KNN][c - 64];
    }
    F[col][c] = v;
  }
  __syncthreads();
  int w = t >> 5, lane = t & 31;
  int lm = lane & 15;
  int kq = (lane < 16) ? 0 : 2;
  for (int mt = w; mt < 12; mt += 8) {
    int mbase = mt * 16;
    v2f a[32];
    #pragma unroll
    for (int j = 0; j < 32; j++) {
      const float* wp = w2 + (size_t)(mbase + lm)*128 + 4*j + kq;
      a[j].x = wp[0]; a[j].y = wp[1];
    }
    for (int ct = 0; ct < 5; ct++) {
      v8f acc = {0.f,0.f,0.f,0.f,0.f,0.f,0.f,0.f};
      int col = ct*16 + lm;
      #pragma unroll
      for (int j = 0; j < 32; j++) {
        v2f bb;
        bb.x = F[col][4*j + kq];
        bb.y = F[col][4*j + kq + 1];
        acc = __builtin_amdgcn_wmma_f32_16x16x4_f32(false, a[j], false, bb,
                                                    (short)0, acc, false, false);
      }
      int rowoff = mbase + ((lane < 16) ? 0 : 8);
      if (MODE == 0) {
        #pragma unroll
        for (int r = 0; r < 8; r++) {
          float sv = acc[r], qv = acc[r]*acc[r];
          #pragma unroll
          for (int msk = 1; msk < 16; msk <<= 1) {
            sv += __shfl_xor(sv, msk, 32);
            qv += __shfl_xor(qv, msk, 32);
          }
          if (lm == 0) {
            int ch = rowoff + r;
            atomicAdd(&sumL[ch], sv); atomicAdd(&sqL[ch], qv);
          }
        }
      } else {
        int pt = col / KNN;
        #pragma unroll
        for (int r = 0; r < 8; r++) {
          int ch = rowoff + r;
          float v = fmaxf(0.f, acc[r]*scale[ch] + shift[ch]);
          atomicMax(&xmaxI[pt*192 + ch], __float_as_int(v));   // v>=0: int order == float order
        }
      }
    }
  }
  __syncthreads();
  if (MODE == 0) {
    if (t < 192) { atomicAdd(&gsum[t], sumL[t]); atomicAdd(&gsq[t], sqL[t]); }
  } else {
    for (int i = t; i < 4*192; i += 256) {
      int pt = i / 192, ch = i % 192;
      xcat[((size_t)(p0 + pt))*256 + 64 + ch] = __int_as_float(xmaxI[i]);
    }
  }
}

// ---------------- conv3: WMMA f32, D(1024 x 16 pts) per WG, K=256 ----------------
template<int MODE>
__global__ __launch_bounds__(256)
void conv3_kernel(const float* __restrict__ xcat, const float* __restrict__ w3,
                  const float* __restrict__ scale, const float* __restrict__ shift,
                  float* __restrict__ gsum, float* __restrict__ gsq,
                  float* __restrict__ hpool) {
  __shared__ float F[16][258];
  int t = threadIdx.x;
  int p0 = blockIdx.x * 16;
  int b = p0 >> 10;                       // 16 pts never cross a batch (1024%16==0)
  for (int i = t; i < 16*256; i += 256) {
    int j = i >> 8, c = i & 255;
    F[j][c] = xcat[((size_t)(p0 + j))*256 + c];
  }
  __syncthreads();
  int w = t >> 5, lane = t & 31;
  int lm = lane & 15;
  int kq = (lane < 16) ? 0 : 2;
  for (int mt = w; mt < 64; mt += 8) {
    int mbase = mt * 16;
    v8f acc = {0.f,0.f,0.f,0.f,0.f,0.f,0.f,0.f};
    for (int kh = 0; kh < 2; kh++) {
      v2f a[32];
      #pragma unroll
      for (int j = 0; j < 32; j++) {
        const float* wp = w3 + (size_t)(mbase + lm)*256 + kh*128 + 4*j + kq;
        a[j].x = wp[0]; a[j].y = wp[1];
      }
      #pragma unroll
      for (int j = 0; j < 32; j++) {
        int k0 = kh*128 + 4*j;
        v2f bb;
        bb.x = F[lm][k0 + kq];
        bb.y = F[lm][k0 + kq + 1];
        acc = __builtin_amdgcn_wmma_f32_16x16x4_f32(false, a[j], false, bb,
                                                    (short)0, acc, false, false);
      }
    }
    int rowoff = mbase + ((lane < 16) ? 0 : 8);
    if (MODE == 0) {
      #pragma unroll
      for (int r = 0; r < 8; r++) {
        float sv = acc[r], qv = acc[r]*acc[r];
        #pragma unroll
        for (int msk = 1; msk < 16; msk <<= 1) {
          sv += __shfl_xor(sv, msk, 32);
          qv += __shfl_xor(qv, msk, 32);
        }
        if (lm == 0) {
          int ch = rowoff + r;
          atomicAdd(&gsum[ch], sv); atomicAdd(&gsq[ch], qv);
        }
      }
    } else {
      #pragma unroll
      for (int r = 0; r < 8; r++) {
        int ch = rowoff + r;
        float v = fmaxf(0.f, acc[r]*scale[ch] + shift[ch]);
        #pragma unroll
        for (int msk = 1; msk < 16; msk <<= 1) v = fmaxf(v, __shfl_xor(v, msk, 32));
        if (lm == 0)
          atomicMax((int*)&hpool[(size_t)b*1024 + ch], __float_as_int(v));
      }
    }
  }
}

// ---------------- FC + LayerNorm + ReLU (B=16, tiny) ----------------
template<int IN, int OUT>
__global__ void fc_ln_kernel(const float* __restrict__ hin, const float* __restrict__ wfc,
                             const float* __restrict__ bfc, const float* __restrict__ g,
                             const float* __restrict__ bb, float* __restrict__ hout) {
  __shared__ float red[OUT];
  __shared__ float red2[OUT];
  int b = blockIdx.x, t = threadIdx.x;     // blockDim == OUT
  const float* hr = hin + (size_t)b*IN;
  const float* wr = wfc + (size_t)t*IN;
  float d = bfc[t];
  for (int c = 0; c < IN; c++) d += wr[c]*hr[c];
  red[t] = d; red2[t] = d*d; __syncthreads();
  for (int s = OUT/2; s > 0; s >>= 1) {
    if (t < s) { red[t] += red[t+s]; red2[t] += red2[t+s]; }
    __syncthreads();
  }
  float mean = red[0] / (float)OUT;
  float var  = red2[0] / (float)OUT - mean*mean;
  float y = (d - mean) * rsqrtf(var + 1e-5f) * g[t] + bb[t];
  hout[(size_t)b*OUT + t] = fmaxf(0.f, y);
}

// ---------------- final linear + log_softmax ----------------
__global__ void out_kernel(const float* __restrict__ h2, const float* __restrict__ wo,
                           const float* __restrict__ bo, float* __restrict__ out) {
  __shared__ float lg[40];
  __shared__ float stat[2];
  int b = blockIdx.x, t = threadIdx.x;     // 64 threads
  if (t < 40) {
    const float* hr = h2 + (size_t)b*256;
    const float* wr = wo + (size_t)t*256;
    float d = bo[t];
    for (int c = 0; c < 256; c++) d += wr[c]*hr[c];
    lg[t] = d;
  }
  __syncthreads();
  if (t == 0) {
    float mx = -3.4e38f;
    for (int i = 0; i < 40; i++) mx = fmaxf(mx, lg[i]);
    float se = 0.f;
    for (int i = 0; i < 40; i++) se += expf(lg[i] - mx);
    stat[0] = mx; stat[1] = logf(se);
  }
  __syncthreads();
  if (t < 40) out[(size_t)b*40 + t] = lg[t] - stat[0] - stat[1];
}

// ---------------- host orchestration ----------------
extern "C" void kernel_launch(void* const* d_in, const int* in_sizes, int n_in,
                              void* d_out, int out_size, void* d_ws, size_t ws_size,
                              hipStream_t stream) {
  const float* x      = (const float*)d_in[0];
  const float* w1     = (const float*)d_in[1];
  const float* bn1g   = (const float*)d_in[2];
  const float* bn1b   = (const float*)d_in[3];
  const float* w2     = (const float*)d_in[4];
  const float* bn2g   = (const float*)d_in[5];
  const float* bn2b   = (const float*)d_in[6];
  const float* w3     = (const float*)d_in[7];
  const float* bn3g   = (const float*)d_in[8];
  const float* bn3b   = (const float*)d_in[9];
  const float* fc1w   = (const float*)d_in[10];
  const float* fc1b   = (const float*)d_in[11];
  const float* ln1g   = (const float*)d_in[12];
  const float* ln1b   = (const float*)d_in[13];
  const float* fc2w   = (const float*)d_in[14];
  const float* fc2b   = (const float*)d_in[15];
  const float* ln2g   = (const float*)d_in[16];
  const float* ln2b   = (const float*)d_in[17];
  const float* outw   = (const float*)d_in[18];
  const float* outb   = (const float*)d_in[19];
  float* out = (float*)d_out;

  // ---- workspace layout (floats) ----
  float* W = (float*)d_ws;
  float*  xp1   = W;                         //  49152
  float*  xp2   = W + 49152;                 // 1048576
  float*  xcat  = W + 1097728;               // 4194304 (cols 0..63 = x1, 64..255 = x2)
  int*    idx1  = (int*)(W + 5292032);       //  327680
  int*    idx2  = (int*)(W + 5619712);       //  327680
  float2* coef2 = (float2*)(W + 5947392);    //  655360 floats
  float*  stats = W + 6602752;               // sum1|sq1|sum2|sq2|sum3|sq3 = 2560
  float*  bnp   = W + 6605312;               // sc1|sh1|sc2|sh2|sc3|sh3   = 2560
  float*  hpool = W + 6607872;               //  16384
  float*  h1    = W + 6624256;               //   8192
  float*  h2    = W + 6632448;               //   4096

  float* sum1 = stats;        float* sq1 = stats + 64;
  float* sum2 = stats + 128;  float* sq2 = stats + 320;
  float* sum3 = stats + 512;  float* sq3 = stats + 1536;
  float* sc1  = bnp;          float* sh1 = bnp + 64;
  float* sc2  = bnp + 128;    float* sh2 = bnp + 320;
  float* sc3  = bnp + 512;    float* sh3 = bnp + 1536;

  // 0) zero per-call accumulators (stats + pooled max)
  zero2_kernel<<<64, 256, 0, stream>>>(stats, 2560, hpool, NPTS_T);

  // 1) layer-1 graph features
  e2p3_kernel<<<(NPTS_T + 255)/256, 256, 0, stream>>>(x, xp1, NPTS_T);
  knn1_kernel<<<NPTS_T, 256, 0, stream>>>(x, idx1);

  // 2) conv1: stats -> BN -> apply (+ReLU+max_k) writes x1 into xcat[:,0:64]
  conv1_kernel<0><<<NPTS_T, 64, 0, stream>>>(xp1, idx1, w1, nullptr, nullptr, sum1, sq1, nullptr);
  bnfin_kernel<<<1, 64, 0, stream>>>(sum1, sq1, bn1g, bn1b, sc1, sh1, 64, 1.f/(float)NEDGE);
  conv1_kernel<1><<<NPTS_T, 64, 0, stream>>>(xp1, idx1, w1, sc1, sh1, nullptr, nullptr, xcat);

  // 3) layer-2 graph features: e2p(x1), knn(x1), mobius coefficients
  e2p64_kernel<<<NPTS_T/8, 256, 0, stream>>>(xcat, 256, xp2, NPTS_T);
  knn2_kernel<<<NPTS_T, 256, 0, stream>>>(xcat, idx2);
  mcoef2_kernel<<<NEDGE/8, 256, 0, stream>>>(xp2, idx2, coef2);

  // 4) conv2 (WMMA): stats -> BN -> apply (+ReLU+max_k) writes x2 into xcat[:,64:256]
  conv2_kernel<0><<<NPTS_T/4, 256, 0, stream>>>(xp2, idx2, coef2, w2, nullptr, nullptr, sum2, sq2, nullptr);
  bnfin_kernel<<<3, 64, 0, stream>>>(sum2, sq2, bn2g, bn2b, sc2, sh2, 192, 1.f/(float)NEDGE);
  conv2_kernel<1><<<NPTS_T/4, 256, 0, stream>>>(xp2, idx2, coef2, w2, sc2, sh2, nullptr, nullptr, xcat);

  // 5) conv3 (WMMA): stats -> BN -> apply (+ReLU+max_N) -> hpool (B,1024)
  conv3_kernel<0><<<NPTS_T/16, 256, 0, stream>>>(xcat, w3, nullptr, nullptr, sum3, sq3, nullptr);
  bnfin_kernel<<<16, 64, 0, stream>>>(sum3, sq3, bn3g, bn3b, sc3, sh3, 1024, 1.f/(float)NPTS_T);
  conv3_kernel<1><<<NPTS_T/16, 256, 0, stream>>>(xcat, w3, sc3, sh3, nullptr, nullptr, hpool);

  // 6) head
  fc_ln_kernel<1024, 512><<<BATCH, 512, 0, stream>>>(hpool, fc1w, fc1b, ln1g, ln1b, h1);
  fc_ln_kernel<512, 256><<<BATCH, 256, 0, stream>>>(h1, fc2w, fc2b, ln2g, ln2b, h2);
  out_kernel<<<BATCH, 64, 0, stream>>>(h2, outw, outb, out);
}